// HybridTokenMixer_31241592111158
// MI455X (gfx1250) — compile-verified
//
#include <hip/hip_runtime.h>
#include <math.h>

typedef __attribute__((ext_vector_type(2))) float v2f;
typedef __attribute__((ext_vector_type(8))) float v8f;

#define CDIV(a,b) (((a)+(b)-1)/(b))

__device__ __forceinline__ float gelu_f(float x) {
  return 0.5f * x * (1.0f + erff(x * 0.70710678118654752440f));
}

// --------------- CDNA5 async global->LDS copy (ASYNCcnt-tracked) -----------
// The gfx1250 builtin takes (global int*, local int*, imm offset, cpol).
// Guarded: host pass / missing builtin falls back to a synchronous copy.
#if __has_builtin(__builtin_amdgcn_global_load_async_to_lds_b32)
typedef __attribute__((address_space(1))) int g_as1_i32;
typedef __attribute__((address_space(3))) int l_as3_i32;
#endif

__device__ __forceinline__ void async_copy_f32(const float* g, float* l) {
#if __has_builtin(__builtin_amdgcn_global_load_async_to_lds_b32)
  __builtin_amdgcn_global_load_async_to_lds_b32(
      (g_as1_i32*)g, (l_as3_i32*)l, /*imm offset*/0, /*cpol*/0);
#else
  *l = *g;
#endif
}

__device__ __forceinline__ void async_wait_all() {
#if __has_builtin(__builtin_amdgcn_global_load_async_to_lds_b32)
#if __has_builtin(__builtin_amdgcn_s_wait_asynccnt)
  __builtin_amdgcn_s_wait_asynccnt(0);
#else
  asm volatile("s_wait_asynccnt 0x0" ::: "memory");
#endif
#endif
}

// ----------------------------- elementwise ---------------------------------
__global__ void k_addgelu(const float* __restrict__ x, const float* __restrict__ add,
                          float* __restrict__ y, int n) {
  int i = blockIdx.x * blockDim.x + threadIdx.x;
  if (i >= n) return;
  float v = x[i];
  if (add) v += add[i];
  y[i] = gelu_f(v);
}

// ----------------------------- 1x1 conv (coalesced on spatial) -------------
__global__ void k_conv1x1(const float* __restrict__ x, const float* __restrict__ w,
                          const float* __restrict__ bias, float* __restrict__ y,
                          int B, int CItot, int cin_start, int CI,
                          int COtot, int cout_start, int CO, int S) {
  int idx = blockIdx.x * blockDim.x + threadIdx.x;
  int total = B * CO * S;
  if (idx >= total) return;
  int s  = idx % S;
  int co = (idx / S) % CO;
  int b  = idx / (S * CO);
  const float* xb = x + ((size_t)b * CItot + cin_start) * S + s;
  const float* wr = w + (size_t)co * CI;
  float acc = bias ? bias[co] : 0.0f;
#pragma unroll 4
  for (int ci = 0; ci < CI; ++ci) acc += wr[ci] * xb[(size_t)ci * S];
  y[((size_t)b * COtot + cout_start + co) * S + s] = acc;
}

// ----------------------------- depthwise conv (odd K, 'same' pad) ----------
__global__ void k_dwconv(const float* __restrict__ x, const float* __restrict__ w,
                         const float* __restrict__ bias, float* __restrict__ y,
                         int B, int CItot, int cin_start, int COtot, int cout_start,
                         int CC, int H, int W, int K) {
  int idx = blockIdx.x * blockDim.x + threadIdx.x;
  int total = B * CC * H * W;
  if (idx >= total) return;
  int wq = idx % W;
  int hp = (idx / W) % H;
  int c  = (idx / (W * H)) % CC;
  int b  = idx / (W * H * CC);
  int p  = K >> 1;
  const float* xb = x + ((size_t)b * CItot + cin_start + c) * H * W;
  const float* wr = w + (size_t)c * K * K;
  float acc = bias ? bias[c] : 0.0f;
  for (int kh = 0; kh < K; ++kh) {
    int ih = hp + kh - p;
    if (ih < 0 || ih >= H) continue;
    for (int kw = 0; kw < K; ++kw) {
      int iw = wq + kw - p;
      if (iw < 0 || iw >= W) continue;
      acc += wr[kh * K + kw] * xb[ih * W + iw];
    }
  }
  y[((size_t)b * COtot + cout_start + c) * H * W + hp * W + wq] = acc;
}

// ------------- dense 3x3 conv on attention maps, LDS-tiled -----------------
// CI fixed at 4; H=W=1024. One block = (b, row hp, 256-wide segment).
// Stages 4ch x 3rows x 258 floats (~12.4 KB) into LDS via async copies,
// then each thread produces CO outputs from conflict-free LDS reads.
__global__ void k_conv3x3_lds(const float* __restrict__ x, const float* __restrict__ w,
                              float* __restrict__ y, int CO) {
  const int CI = 4, H = 1024, W = 1024, SEG = 256, LW = 264; // LW: padded row
  __shared__ float xs[CI * 3 * LW];
  int tid = threadIdx.x;
  int wt = blockIdx.x & 3;
  int hp = (blockIdx.x >> 2) & 1023;
  int b  = blockIdx.x >> 12;
  int w0 = wt * SEG;

  // fill halo tile: dx in [0,258) maps to iw = w0 + dx - 1
  for (int t = tid; t < CI * 3 * (SEG + 2); t += 256) {
    int dx = t % (SEG + 2);
    int dy = (t / (SEG + 2)) % 3;
    int ci = t / ((SEG + 2) * 3);
    int ih = hp + dy - 1;
    int iw = w0 + dx - 1;
    float* dst = &xs[(ci * 3 + dy) * LW + dx];
    if (ih >= 0 && ih < H && iw >= 0 && iw < W) {
      async_copy_f32(x + ((size_t)b * CI + ci) * H * W + (size_t)ih * W + iw, dst);
    } else {
      *dst = 0.0f;
    }
  }
  async_wait_all();
  __syncthreads();

  float acc[4] = {0.0f, 0.0f, 0.0f, 0.0f};
  for (int ci = 0; ci < CI; ++ci) {
    const float* r0 = &xs[(ci * 3 + 0) * LW + tid];
    const float* r1 = &xs[(ci * 3 + 1) * LW + tid];
    const float* r2 = &xs[(ci * 3 + 2) * LW + tid];
    float t0 = r0[0], t1 = r0[1], t2 = r0[2];
    float t3 = r1[0], t4 = r1[1], t5 = r1[2];
    float t6 = r2[0], t7 = r2[1], t8 = r2[2];
    for (int co = 0; co < CO; ++co) {
      const float* wr = w + ((size_t)co * CI + ci) * 9;  // uniform -> scalar loads
      acc[co] += wr[0] * t0 + wr[1] * t1 + wr[2] * t2
               + wr[3] * t3 + wr[4] * t4 + wr[5] * t5
               + wr[6] * t6 + wr[7] * t7 + wr[8] * t8;
    }
  }
  for (int co = 0; co < CO; ++co)
    y[((size_t)b * CO + co) * H * W + (size_t)hp * W + w0 + tid] = acc[co];
}

// ----------------------------- BN: deterministic 2-phase stats -------------
__global__ void k_bn_reduce(const float* __restrict__ x, float* __restrict__ mean_out,
                            float* __restrict__ rstd_out,
                            int B, int Ctot, int cstart, long long S, float eps) {
  int c   = blockIdx.x;
  int tid = threadIdx.x;
  __shared__ double ssum[256];
  __shared__ double ssq[256];
  double s = 0.0, q = 0.0;
  long long per = (long long)B * S;
  for (long long i = tid; i < per; i += blockDim.x) {
    long long b  = i / S;
    long long sp = i % S;
    float v = x[((size_t)b * Ctot + cstart + c) * S + sp];
    s += (double)v;
    q += (double)v * (double)v;
  }
  ssum[tid] = s; ssq[tid] = q;
  __syncthreads();
  for (int off = 128; off > 0; off >>= 1) {
    if (tid < off) { ssum[tid] += ssum[tid + off]; ssq[tid] += ssq[tid + off]; }
    __syncthreads();
  }
  if (tid == 0) {
    double m   = ssum[0] / (double)per;
    double var = ssq[0] / (double)per - m * m;
    mean_out[c] = (float)m;
    rstd_out[c] = (float)(1.0 / sqrt(var + (double)eps));
  }
}

__global__ void k_bn_apply(const float* __restrict__ x, const float* __restrict__ g,
                           const float* __restrict__ bta, const float* __restrict__ mean,
                           const float* __restrict__ rstd, const float* __restrict__ residual,
                           float* __restrict__ y,
                           int B, int Ctot, int cstart, int CC, long long S, int gelu_after) {
  long long idx   = (long long)blockIdx.x * blockDim.x + threadIdx.x;
  long long total = (long long)B * CC * S;
  if (idx >= total) return;
  long long sp = idx % S;
  int c = (int)((idx / S) % CC);
  int b = (int)(idx / (S * CC));
  size_t pos = ((size_t)b * Ctot + cstart + c) * S + sp;
  float v = (x[pos] - mean[c]) * rstd[c] * g[c] + bta[c];
  if (gelu_after) v = gelu_f(v);
  if (residual) v += residual[pos];
  y[pos] = v;
}

// ----------------------------- attn * softmax(attn, -1), row = 1024 --------
__global__ void k_softmax_mul(const float* __restrict__ x, float* __restrict__ y) {
  const int L = 1024, T = 256;
  int r   = blockIdx.x;
  int tid = threadIdx.x;
  const float* xr = x + (size_t)r * L;
  float* yr = y + (size_t)r * L;
  float v[4];
  float mx = -INFINITY;
#pragma unroll
  for (int t = 0; t < 4; ++t) { v[t] = xr[tid + t * T]; mx = fmaxf(mx, v[t]); }
  __shared__ float sm[256];
  sm[tid] = mx;
  __syncthreads();
  for (int off = 128; off > 0; off >>= 1) {
    if (tid < off) sm[tid] = fmaxf(sm[tid], sm[tid + off]);
    __syncthreads();
  }
  mx = sm[0];
  __syncthreads();
  float e[4], se = 0.0f;
#pragma unroll
  for (int t = 0; t < 4; ++t) { e[t] = expf(v[t] - mx); se += e[t]; }
  sm[tid] = se;
  __syncthreads();
  for (int off = 128; off > 0; off >>= 1) {
    if (tid < off) sm[tid] += sm[tid + off];
    __syncthreads();
  }
  float inv = 1.0f / sm[0];
#pragma unroll
  for (int t = 0; t < 4; ++t) yr[tid + t * T] = v[t] * e[t] * inv;
}

// ----------------------------- WMMA fp32: attn = scale * q^T k -------------
// Q,K stored (32 be, 128 c, 1024 n) K-major; attn (32, 1024, 1024).
// All 8 waves of a block share one Q strip (16 i x 128 c = 8 KB): it is
// staged once into LDS via async copies; B fragments stream from global.
__global__ void k_qk_wmma(const float* __restrict__ Q, const float* __restrict__ Km,
                          float* __restrict__ attn, float scale) {
  __shared__ float qs[128 * 16];  // qs[c*16 + i] = Q[c][i0 + i]
  int tid  = threadIdx.x;
  int lane = tid & 31;
  int lo = lane & 15, hi = lane >> 4;

  int wave0  = blockIdx.x * 8;          // block-uniform wave base
  int itile  = (wave0 >> 4) & 63;       // same for all 8 waves in block
  int be     = wave0 >> 10;
  int i0     = itile * 16;
  const float* Qb = Q + (size_t)be * 128 * 1024;

  // cooperative async stage of the Q strip: 2048 dwords, 8 per thread
  for (int t = tid; t < 2048; t += 256) {
    int c = t >> 4, i = t & 15;
    async_copy_f32(Qb + (size_t)c * 1024 + i0 + i, &qs[t]);
  }
  async_wait_all();
  __syncthreads();

  int wave   = wave0 + (tid >> 5);
  int jstrip = wave & 15;
  int j0     = jstrip * 64;
  const float* Kb = Km + (size_t)be * 128 * 1024;

  v8f acc0 = {}, acc1 = {}, acc2 = {}, acc3 = {};
  for (int k0 = 0; k0 < 128; k0 += 4) {
    int row = k0 + 2 * hi;
    v2f a;
    a.x = qs[row * 16 + lo];
    a.y = qs[(row + 1) * 16 + lo];
    const float* K0 = Kb + (size_t)row * 1024 + j0 + lo;
    const float* K1 = K0 + 1024;
    v2f b0, b1, b2, b3;
    b0.x = K0[0];  b0.y = K1[0];
    b1.x = K0[16]; b1.y = K1[16];
    b2.x = K0[32]; b2.y = K1[32];
    b3.x = K0[48]; b3.y = K1[48];
    acc0 = __builtin_amdgcn_wmma_f32_16x16x4_f32(false, a, false, b0, (short)0, acc0, false, false);
    acc1 = __builtin_amdgcn_wmma_f32_16x16x4_f32(false, a, false, b1, (short)0, acc1, false, false);
    acc2 = __builtin_amdgcn_wmma_f32_16x16x4_f32(false, a, false, b2, (short)0, acc2, false, false);
    acc3 = __builtin_amdgcn_wmma_f32_16x16x4_f32(false, a, false, b3, (short)0, acc3, false, false);
  }
  float* out = attn + ((size_t)be * 1024 + i0 + 8 * hi) * 1024 + j0 + lo;
#pragma unroll
  for (int r = 0; r < 8; ++r) {
    size_t ro = (size_t)r * 1024;
    out[ro + 0]  = acc0[r] * scale;
    out[ro + 16] = acc1[r] * scale;
    out[ro + 32] = acc2[r] * scale;
    out[ro + 48] = acc3[r] * scale;
  }
}

// ----------------------------- WMMA fp32: out2 = attn3 @ v^T ---------------
// A3 (8,1024,1024) row-major; V (8,128,1024); writes xc channels [128,256).
// D[i][c] = sum_j A3[i][j] * V[c][j]; fragment pairs are contiguous in j.
__global__ void k_av_wmma(const float* __restrict__ A3, const float* __restrict__ V,
                          float* __restrict__ xc) {
  int wave = blockIdx.x * (blockDim.x >> 5) + (threadIdx.x >> 5);
  int lane = threadIdx.x & 31;
  int lo = lane & 15, hi = lane >> 4;
  int cstrip = wave & 1;
  int itile  = (wave >> 1) & 63;
  int b      = wave >> 7;
  int i0 = itile * 16, c0 = cstrip * 64;
  const float* Ab = A3 + (size_t)b * 1024 * 1024;
  const float* Vb = V  + (size_t)b * 128 * 1024;
  v8f acc0 = {}, acc1 = {}, acc2 = {}, acc3 = {};
  for (int k0 = 0; k0 < 1024; k0 += 4) {
    int row = k0 + 2 * hi;                       // even -> 8B aligned v2f loads
    v2f a  = *(const v2f*)(Ab + (size_t)(i0 + lo) * 1024 + row);
    v2f b0 = *(const v2f*)(Vb + (size_t)(c0 + lo) * 1024 + row);
    v2f b1 = *(const v2f*)(Vb + (size_t)(c0 + 16 + lo) * 1024 + row);
    v2f b2 = *(const v2f*)(Vb + (size_t)(c0 + 32 + lo) * 1024 + row);
    v2f b3 = *(const v2f*)(Vb + (size_t)(c0 + 48 + lo) * 1024 + row);
    acc0 = __builtin_amdgcn_wmma_f32_16x16x4_f32(false, a, false, b0, (short)0, acc0, false, false);
    acc1 = __builtin_amdgcn_wmma_f32_16x16x4_f32(false, a, false, b1, (short)0, acc1, false, false);
    acc2 = __builtin_amdgcn_wmma_f32_16x16x4_f32(false, a, false, b2, (short)0, acc2, false, false);
    acc3 = __builtin_amdgcn_wmma_f32_16x16x4_f32(false, a, false, b3, (short)0, acc3, false, false);
  }
#pragma unroll
  for (int r = 0; r < 8; ++r) {
    int i = i0 + r + 8 * hi;
    xc[((size_t)b * 256 + 128 + c0 + 0  + lo) * 1024 + i] = acc0[r];
    xc[((size_t)b * 256 + 128 + c0 + 16 + lo) * 1024 + i] = acc1[r];
    xc[((size_t)b * 256 + 128 + c0 + 32 + lo) * 1024 + i] = acc2[r];
    xc[((size_t)b * 256 + 128 + c0 + 48 + lo) * 1024 + i] = acc3[r];
  }
}

// ============================================================================
extern "C" void kernel_launch(void* const* d_in, const int* in_sizes, int n_in,
                              void* d_out, int out_size, void* d_ws, size_t ws_size,
                              hipStream_t stream) {
  (void)in_sizes; (void)n_in; (void)out_size; (void)ws_size;
  const float* x       = (const float*)d_in[0];
  const float* q_dw_w  = (const float*)d_in[1];
  const float* q_dw_b  = (const float*)d_in[2];
  const float* q_pw_w  = (const float*)d_in[3];
  const float* q_pw_b  = (const float*)d_in[4];
  const float* k_dw_w  = (const float*)d_in[5];
  const float* k_dw_b  = (const float*)d_in[6];
  const float* k_pw_w  = (const float*)d_in[7];
  const float* k_pw_b  = (const float*)d_in[8];
  const float* v_dw_w  = (const float*)d_in[9];
  const float* v_dw_b  = (const float*)d_in[10];
  const float* v_pw_w  = (const float*)d_in[11];
  const float* v_pw_b  = (const float*)d_in[12];
  const float* a0_w    = (const float*)d_in[13];
  const float* a0_g    = (const float*)d_in[14];
  const float* a0_b    = (const float*)d_in[15];
  const float* a1_w    = (const float*)d_in[16];
  const float* fc1_w   = (const float*)d_in[17];
  const float* fc1_g   = (const float*)d_in[18];
  const float* fc1_b   = (const float*)d_in[19];
  const float* ms_w[4] = {(const float*)d_in[20], (const float*)d_in[22],
                          (const float*)d_in[24], (const float*)d_in[26]};
  const float* ms_b[4] = {(const float*)d_in[21], (const float*)d_in[23],
                          (const float*)d_in[25], (const float*)d_in[27]};
  const float* n1_g    = (const float*)d_in[28];
  const float* n1_b    = (const float*)d_in[29];
  const float* fc2_w   = (const float*)d_in[30];
  const float* fc2_g   = (const float*)d_in[31];
  const float* fc2_b   = (const float*)d_in[32];
  const float* p_dw_w  = (const float*)d_in[33];
  const float* p_dw_b  = (const float*)d_in[34];
  const float* p_bn1_g = (const float*)d_in[35];
  const float* p_bn1_b = (const float*)d_in[36];
  const float* p_pw1_w = (const float*)d_in[37];
  const float* p_pw1_b = (const float*)d_in[38];
  const float* p_bn2_g = (const float*)d_in[39];
  const float* p_bn2_b = (const float*)d_in[40];
  const float* p_pw2_w = (const float*)d_in[41];
  const float* p_pw2_b = (const float*)d_in[42];
  const float* p_bn3_g = (const float*)d_in[43];
  const float* p_bn3_b = (const float*)d_in[44];

  float* out = (float*)d_out;
  const int B = 8, S = 1024, HH = 32, WW = 32;
  const float eps = 1e-5f;

  // -------- workspace layout (floats) --------
  float* ws = (float*)d_ws;
  size_t off = 0;
  float* h     = ws + off; off += (size_t)B * 32 * S;
  float* ms    = ws + off; off += (size_t)B * 32 * S;
  float* dtmp  = ws + off; off += (size_t)B * 128 * S;
  float* qb    = ws + off; off += (size_t)B * 512 * S;
  float* kb    = ws + off; off += (size_t)B * 512 * S;
  float* vb    = ws + off; off += (size_t)B * 128 * S;
  float* attnA = ws + off; off += (size_t)B * 4 * 1024 * 1024;
  float* attnB = ws + off; off += (size_t)B * 4 * 1024 * 1024;
  float* a3    = ws + off; off += (size_t)B * 1024 * 1024;
  float* xc    = ws + off; off += (size_t)B * 256 * S;
  float* y1    = ws + off; off += (size_t)B * 256 * S;
  float* y2    = ws + off; off += (size_t)B * 32 * S;
  float* y3    = ws + off; off += (size_t)B * 256 * S;
  float* meanb = ws + off; off += 256;
  float* rstdb = ws + off; off += 256;

  const int T = 256;

  // ================= Branch 1: conv MLP on x1 (channels 0..127) ============
  k_conv1x1<<<CDIV(B*32*S, T), T, 0, stream>>>(x, fc1_w, nullptr, h, B, 256, 0, 128, 32, 0, 32, S);
  k_addgelu<<<CDIV(B*32*S, T), T, 0, stream>>>(h, nullptr, h, B*32*S);
  k_bn_reduce<<<32, T, 0, stream>>>(h, meanb, rstdb, B, 32, 0, S, eps);
  k_bn_apply<<<CDIV(B*32*S, T), T, 0, stream>>>(h, fc1_g, fc1_b, meanb, rstdb, nullptr, h,
                                                B, 32, 0, 32, S, 0);
  {
    const int ksz[4] = {1, 3, 5, 7};
    for (int p = 0; p < 4; ++p) {
      k_dwconv<<<CDIV(B*8*S, T), T, 0, stream>>>(h, ms_w[p], ms_b[p], ms,
                                                 B, 32, 8*p, 32, 8*p, 8, HH, WW, ksz[p]);
    }
  }
  k_addgelu<<<CDIV(B*32*S, T), T, 0, stream>>>(ms, h, ms, B*32*S);
  k_bn_reduce<<<32, T, 0, stream>>>(ms, meanb, rstdb, B, 32, 0, S, eps);
  k_bn_apply<<<CDIV(B*32*S, T), T, 0, stream>>>(ms, n1_g, n1_b, meanb, rstdb, nullptr, ms,
                                                B, 32, 0, 32, S, 0);
  k_conv1x1<<<CDIV(B*128*S, T), T, 0, stream>>>(ms, fc2_w, nullptr, xc, B, 32, 0, 32, 256, 0, 128, S);
  k_bn_reduce<<<128, T, 0, stream>>>(xc, meanb, rstdb, B, 256, 0, S, eps);
  k_bn_apply<<<CDIV(B*128*S, T), T, 0, stream>>>(xc, fc2_g, fc2_b, meanb, rstdb, nullptr, xc,
                                                 B, 256, 0, 128, S, 0);

  // ================= Branch 2: attention on x2 (channels 128..255) =========
  k_dwconv<<<CDIV(B*128*S, T), T, 0, stream>>>(x, q_dw_w, q_dw_b, dtmp, B, 256, 128, 128, 0, 128, HH, WW, 3);
  k_conv1x1<<<CDIV(B*512*S, T), T, 0, stream>>>(dtmp, q_pw_w, q_pw_b, qb, B, 128, 0, 128, 512, 0, 512, S);
  k_dwconv<<<CDIV(B*128*S, T), T, 0, stream>>>(x, k_dw_w, k_dw_b, dtmp, B, 256, 128, 128, 0, 128, HH, WW, 3);
  k_conv1x1<<<CDIV(B*512*S, T), T, 0, stream>>>(dtmp, k_pw_w, k_pw_b, kb, B, 128, 0, 128, 512, 0, 512, S);
  k_dwconv<<<CDIV(B*128*S, T), T, 0, stream>>>(x, v_dw_w, v_dw_b, dtmp, B, 256, 128, 128, 0, 128, HH, WW, 3);
  k_conv1x1<<<CDIV(B*128*S, T), T, 0, stream>>>(dtmp, v_pw_w, v_pw_b, vb, B, 128, 0, 128, 128, 0, 128, S);

  // attn0 = scale * q^T k   (WMMA fp32, LDS-staged A) -> attnA
  float scale = 1.0f / sqrtf(128.0f);
  k_qk_wmma<<<4096, 256, 0, stream>>>(qb, kb, attnA, scale);

  // attn1 = bn(conv3x3(attn0)) -> attnB (bn in place)
  k_conv3x3_lds<<<32768, 256, 0, stream>>>(attnA, a0_w, attnB, 4);
  k_bn_reduce<<<4, T, 0, stream>>>(attnB, meanb, rstdb, 8, 4, 0, (long long)1024*1024, eps);
  k_bn_apply<<<CDIV(8*4*1024*1024, T), T, 0, stream>>>(attnB, a0_g, a0_b, meanb, rstdb, nullptr,
                                                       attnB, 8, 4, 0, 4, (long long)1024*1024, 0);
  // attn2 = attn1 * softmax(attn1, -1) -> attnA
  k_softmax_mul<<<8*4*1024, 256, 0, stream>>>(attnB, attnA);
  // attn3 = conv3x3(attn2, a1_w) 4->1 -> a3
  k_conv3x3_lds<<<32768, 256, 0, stream>>>(attnA, a1_w, a3, 1);
  // out2 = attn3 @ v (WMMA fp32), written into xc channels [128,256)
  k_av_wmma<<<128, 256, 0, stream>>>(a3, vb, xc);

  // ================= Projection + residual =================================
  k_dwconv<<<CDIV(B*256*S, T), T, 0, stream>>>(xc, p_dw_w, p_dw_b, y1, B, 256, 0, 256, 0, 256, HH, WW, 3);
  k_bn_reduce<<<256, T, 0, stream>>>(y1, meanb, rstdb, B, 256, 0, S, eps);
  k_bn_apply<<<CDIV(B*256*S, T), T, 0, stream>>>(y1, p_bn1_g, p_bn1_b, meanb, rstdb, nullptr, y1,
                                                 B, 256, 0, 256, S, 1);
  k_conv1x1<<<CDIV(B*32*S, T), T, 0, stream>>>(y1, p_pw1_w, p_pw1_b, y2, B, 256, 0, 256, 32, 0, 32, S);
  k_bn_reduce<<<32, T, 0, stream>>>(y2, meanb, rstdb, B, 32, 0, S, eps);
  k_bn_apply<<<CDIV(B*32*S, T), T, 0, stream>>>(y2, p_bn2_g, p_bn2_b, meanb, rstdb, nullptr, y2,
                                                B, 32, 0, 32, S, 1);
  k_conv1x1<<<CDIV(B*256*S, T), T, 0, stream>>>(y2, p_pw2_w, p_pw2_b, y3, B, 32, 0, 32, 256, 0, 256, S);
  k_bn_reduce<<<256, T, 0, stream>>>(y3, meanb, rstdb, B, 256, 0, S, eps);
  k_bn_apply<<<CDIV(B*256*S, T), T, 0, stream>>>(y3, p_bn3_g, p_bn3_b, meanb, rstdb, xc, out,
                                                 B, 256, 0, 256, S, 0);
}